// GraphSAGEModel_89069031784798
// MI455X (gfx1250) — compile-verified
//
#include <hip/hip_runtime.h>

// GraphSAGE (2x SAGEConv mean-agg + linear classifier) for MI455X / gfx1250.
// fp32 end-to-end; matmuls via V_WMMA_F32_16X16X4_F32 (exact fp32 path).
// Edge aggregation via global_atomic_add_f32 (working set fits 192MB L2).
// Each GEMM wave owns a 16-node x 64-output tile: A loaded once per k-step,
// 4 WMMAs issued against 4 independent accumulators (better XDL overlap,
// 4x less A traffic than one-N-tile-per-wave).

typedef __attribute__((ext_vector_type(2))) float v2f;
typedef __attribute__((ext_vector_type(8))) float v8f;

#define N_NODES   100000
#define N_EDGES   3200000
#define IN_CH     24
#define HIDDEN    64
#define N_CLASSES 16
#define NODES_PAD 100352  // N_NODES rounded up to 1024 (keeps 8B alignment of regions)

// ---------------------------------------------------------------- zero scratch
__global__ void zero_kernel(float* __restrict__ p, long long n) {
    long long i = (long long)blockIdx.x * blockDim.x + threadIdx.x;
    if (i < n) p[i] = 0.0f;
}

// ------------------------------------------- layer-1 scatter: agg1 += x[src], deg
// One wave per edge; lane = input channel (24 active lanes). Coalesced atomics.
__global__ void scatter1_kernel(const float* __restrict__ x,
                                const int* __restrict__ src,
                                const int* __restrict__ dst,
                                float* __restrict__ agg1,
                                float* __restrict__ deg) {
    long long tid = (long long)blockIdx.x * blockDim.x + threadIdx.x;
    long long e = tid >> 5;
    int c = (int)(tid & 31);
    if (e >= N_EDGES) return;
    int s = src[e];
    int d = dst[e];
    if (c == 0) atomicAdd(&deg[d], 1.0f);
    if (c < IN_CH)
        atomicAdd(&agg1[(long long)d * IN_CH + c], x[(long long)s * IN_CH + c]);
}

// ------------------------------------------------------- inverse degree (>=1)
__global__ void invdeg_kernel(const float* __restrict__ deg,
                              float* __restrict__ invd) {
    int i = blockIdx.x * blockDim.x + threadIdx.x;
    if (i < N_NODES) invd[i] = 1.0f / fmaxf(deg[i], 1.0f);
}

// ------------------------------------------- layer-2 scatter: agg2 += h[src]
// Thread per (edge, channel): each wave covers 32 contiguous channels -> 128B
// coalesced loads and atomic adds.
__global__ void scatter2_kernel(const float* __restrict__ h,
                                const int* __restrict__ src,
                                const int* __restrict__ dst,
                                float* __restrict__ agg2) {
    long long tid = (long long)blockIdx.x * blockDim.x + threadIdx.x;
    long long e = tid >> 6;            // HIDDEN == 64
    int c = (int)(tid & 63);
    if (e >= N_EDGES) return;
    atomicAdd(&agg2[(long long)dst[e] * HIDDEN + c],
              h[(long long)src[e] * HIDDEN + c]);
}

// -------------------------------------------------------------- SAGE layer GEMM
// out[m,n] = act( sum_k mean[m,k]*Wl[n,k] + bias[n] + sum_k self[m,k]*Wr[n,k] )
// mean[m,k] = agg[m,k] * invd[m];  NOUT = HIDDEN = 64 (4 N-tiles per wave).
// V_WMMA_F32_16X16X4_F32 operand layouts (ISA 7.12.2):
//   A 16x4: lanes 0-15 hold row M=lane with {v0,v1}={K,K+1};
//           lanes 16-31 hold row M=lane-16 with {v0,v1}={K+2,K+3}.
//   B 4x16 mirrors A (N on lanes, K per VGPR).
//   C/D:    v[i] = element (M = i + 8*(lane>>4), N = lane&15).
template <int K, bool RELU>
__global__ void sage_wmma_kernel(const float* __restrict__ agg,   // [N_NODES,K]
                                 const float* __restrict__ invd,  // [N_NODES]
                                 const float* __restrict__ self,  // [N_NODES,K]
                                 const float* __restrict__ Wl,    // [HIDDEN,K]
                                 const float* __restrict__ bias,  // [HIDDEN]
                                 const float* __restrict__ Wr,    // [HIDDEN,K]
                                 float* __restrict__ out) {       // [N_NODES,HIDDEN]
    const int lane = threadIdx.x & 31;
    const int wave = blockIdx.x * (blockDim.x >> 5) + (threadIdx.x >> 5);
    if (wave >= N_NODES / 16) return;

    const int mrow = (wave << 4) + (lane & 15);   // node row this lane feeds A with
    const int koff = (lane >> 4) << 1;            // 0 (lanes 0-15) or 2 (lanes 16-31)
    const int nrow = lane & 15;                   // B column within an N-tile

    const float scale = invd[mrow];
    v8f acc[4] = {v8f{}, v8f{}, v8f{}, v8f{}};

    // mean-aggregate @ Wl^T : one A load feeds 4 independent WMMAs
#pragma unroll
    for (int k = 0; k < K; k += 4) {
        v2f a;
        a.x = agg[(long long)mrow * K + k + koff]     * scale;
        a.y = agg[(long long)mrow * K + k + koff + 1] * scale;
#pragma unroll
        for (int t = 0; t < 4; ++t) {
            const int ncol = (t << 4) + nrow;
            v2f b;
            b.x = Wl[ncol * K + k + koff];
            b.y = Wl[ncol * K + k + koff + 1];
            acc[t] = __builtin_amdgcn_wmma_f32_16x16x4_f32(
                false, a, false, b, (short)0, acc[t], false, false);
        }
    }
    // self @ Wr^T (same accumulators)
#pragma unroll
    for (int k = 0; k < K; k += 4) {
        v2f a;
        a.x = self[(long long)mrow * K + k + koff];
        a.y = self[(long long)mrow * K + k + koff + 1];
#pragma unroll
        for (int t = 0; t < 4; ++t) {
            const int ncol = (t << 4) + nrow;
            v2f b;
            b.x = Wr[ncol * K + k + koff];
            b.y = Wr[ncol * K + k + koff + 1];
            acc[t] = __builtin_amdgcn_wmma_f32_16x16x4_f32(
                false, a, false, b, (short)0, acc[t], false, false);
        }
    }

    const int mbase = (wave << 4) + ((lane >> 4) << 3);
#pragma unroll
    for (int t = 0; t < 4; ++t) {
        const int n    = (t << 4) + nrow;
        const float bn = bias[n];
#pragma unroll
        for (int i = 0; i < 8; ++i) {
            float v = acc[t][i] + bn;
            if (RELU) v = fmaxf(v, 0.0f);
            out[(long long)(mbase + i) * HIDDEN + n] = v;
        }
    }
}

// ---------------------------------------------------------- classifier GEMM
// logits[m,n] = sum_k emb[m,k]*Wc[n,k] + bc[n]; K=64, NOUT=16 (one N tile).
__global__ void classifier_wmma_kernel(const float* __restrict__ emb,  // [N,64]
                                       const float* __restrict__ Wc,   // [16,64]
                                       const float* __restrict__ bc,   // [16]
                                       float* __restrict__ logits) {   // [N,16]
    const int lane = threadIdx.x & 31;
    const int wave = blockIdx.x * (blockDim.x >> 5) + (threadIdx.x >> 5);
    if (wave >= N_NODES / 16) return;

    const int mrow = (wave << 4) + (lane & 15);
    const int koff = (lane >> 4) << 1;
    const int ncol = lane & 15;

    v8f acc = {};
#pragma unroll
    for (int k = 0; k < HIDDEN; k += 4) {
        v2f a, b;
        a.x = emb[(long long)mrow * HIDDEN + k + koff];
        a.y = emb[(long long)mrow * HIDDEN + k + koff + 1];
        b.x = Wc[ncol * HIDDEN + k + koff];
        b.y = Wc[ncol * HIDDEN + k + koff + 1];
        acc = __builtin_amdgcn_wmma_f32_16x16x4_f32(false, a, false, b,
                                                    (short)0, acc, false, false);
    }

    const float bn  = bc[ncol];
    const int mbase = (wave << 4) + ((lane >> 4) << 3);
#pragma unroll
    for (int i = 0; i < 8; ++i)
        logits[(long long)(mbase + i) * N_CLASSES + ncol] = acc[i] + bn;
}

// ---------------------------------------------------------------------- launch
extern "C" void kernel_launch(void* const* d_in, const int* in_sizes, int n_in,
                              void* d_out, int out_size, void* d_ws, size_t ws_size,
                              hipStream_t stream) {
    const float* x    = (const float*)d_in[0];
    const int*   ei   = (const int*)d_in[1];    // [2, N_EDGES]
    const float* W1l  = (const float*)d_in[2];  // [64,24]
    const float* b1   = (const float*)d_in[3];  // [64]
    const float* W1r  = (const float*)d_in[4];  // [64,24]
    const float* W2l  = (const float*)d_in[5];  // [64,64]
    const float* b2   = (const float*)d_in[6];  // [64]
    const float* W2r  = (const float*)d_in[7];  // [64,64]
    const float* Wc   = (const float*)d_in[8];  // [16,64]
    const float* bc   = (const float*)d_in[9];  // [16]
    const int* src = ei;
    const int* dst = ei + N_EDGES;

    // Workspace layout (floats): deg | invd | agg1 | agg2 | h   (~61.6 MB)
    float* ws   = (float*)d_ws;
    float* deg  = ws;
    float* invd = ws + NODES_PAD;
    float* agg1 = ws + 2 * (long long)NODES_PAD;
    float* agg2 = agg1 + (long long)N_NODES * IN_CH;
    float* h    = agg2 + (long long)N_NODES * HIDDEN;

    // Output layout: logits [N,16] then embeddings [N,64]
    float* logits = (float*)d_out;
    float* emb    = logits + (long long)N_NODES * N_CLASSES;

    // 1) zero deg/invd/agg1/agg2 (h is fully overwritten)
    long long nzero = 2LL * NODES_PAD + (long long)N_NODES * IN_CH
                    + (long long)N_NODES * HIDDEN;
    zero_kernel<<<(unsigned)((nzero + 255) / 256), 256, 0, stream>>>(ws, nzero);

    // 2) layer-1 edge scatter (+degree): one wave per edge
    {
        long long threads = (long long)N_EDGES * 32;
        scatter1_kernel<<<(unsigned)((threads + 255) / 256), 256, 0, stream>>>(
            x, src, dst, agg1, deg);
    }

    // 3) inverse degree
    invdeg_kernel<<<(N_NODES + 255) / 256, 256, 0, stream>>>(deg, invd);

    // 4) layer 1: h = relu(mean@W1l^T + b1 + x@W1r^T)  [WMMA, 16x64 tile/wave]
    {
        int waves  = N_NODES / 16;
        int blocks = (waves + 7) / 8;
        sage_wmma_kernel<IN_CH, true>
            <<<blocks, 256, 0, stream>>>(agg1, invd, x, W1l, b1, W1r, h);
    }

    // 5) layer-2 edge scatter: thread per (edge, channel)
    {
        long long threads = (long long)N_EDGES * HIDDEN;
        scatter2_kernel<<<(unsigned)((threads + 255) / 256), 256, 0, stream>>>(
            h, src, dst, agg2);
    }

    // 6) layer 2: emb = mean@W2l^T + b2 + h@W2r^T  [WMMA] -> d_out tail
    {
        int waves  = N_NODES / 16;
        int blocks = (waves + 7) / 8;
        sage_wmma_kernel<HIDDEN, false>
            <<<blocks, 256, 0, stream>>>(agg2, invd, h, W2l, b2, W2r, emb);
    }

    // 7) classifier: logits = emb@Wc^T + bc  [WMMA] -> d_out head
    {
        int waves  = N_NODES / 16;
        int blocks = (waves + 7) / 8;
        classifier_wmma_kernel<<<blocks, 256, 0, stream>>>(emb, Wc, bc, logits);
    }
}